// ProjectNet_63522566307880
// MI455X (gfx1250) — compile-verified
//
#include <hip/hip_runtime.h>

typedef __attribute__((ext_vector_type(2))) float v2f;
typedef __attribute__((ext_vector_type(8))) float v8f;

#define D_DIM 1024
#define B_DIM 512
#define MC_DIM 256
#define TOLV 0.01f
#define NELEM (B_DIM * D_DIM)

// ---------------------------------------------------------------------------
// Generic NT GEMM: C[M,N] = epilogue( sum_k Aop[i,k] * Bmat[j,k] )
// Block = 128 threads (4 waves), block tile 64x64, wave tile 32x32
// (2x2 fragments of V_WMMA_F32_16X16X4_F32).
// Fragment layouts per CDNA5 ISA 7.12.2 (f32, 16x16x4):
//   A 16x4 : lane l in [0,15] holds row l; VGPR0/1 = K = 2*(lane>>4)+{0,1}
//   B 4x16 : lane l holds col (l&15);     VGPR0/1 = K = 2*(lane>>4)+{0,1}
//   C 16x16: VGPR r: lanes0-15 -> (M=r, N=lane); lanes16-31 -> (M=8+r)
// modes: 0: C=acc   1: C=acc-aux1[col]   2: C=aux1[o]-acc
//        3: C=aux1[o] - (s1*acc + s2*aux2[o])
// ---------------------------------------------------------------------------
__global__ __launch_bounds__(128) void k_gemm_nt(
    const float* __restrict__ Aop, const float* __restrict__ Bmat,
    float* __restrict__ Cout, int K, int lda, int ldb, int ldc,
    int mode, const float* __restrict__ aux1, const float* __restrict__ aux2,
    float s1, float s2)
{
    const int lane = threadIdx.x & 31;
    const int wave = threadIdx.x >> 5;
    const int l = lane & 15;
    const int h = lane >> 4;
    const int rowBase = blockIdx.y * 64 + (wave >> 1) * 32;
    const int colBase = blockIdx.x * 64 + (wave & 1) * 32;

    const float* a0p = Aop + (size_t)(rowBase + l) * lda + 2 * h;
    const float* a1p = Aop + (size_t)(rowBase + 16 + l) * lda + 2 * h;
    const float* b0p = Bmat + (size_t)(colBase + l) * ldb + 2 * h;
    const float* b1p = Bmat + (size_t)(colBase + 16 + l) * ldb + 2 * h;

    v8f c00 = {}, c01 = {}, c10 = {}, c11 = {};
#pragma unroll 4
    for (int k = 0; k < K; k += 4) {
        v2f a0 = *(const v2f*)(a0p + k);
        v2f a1 = *(const v2f*)(a1p + k);
        v2f b0 = *(const v2f*)(b0p + k);
        v2f b1 = *(const v2f*)(b1p + k);
        c00 = __builtin_amdgcn_wmma_f32_16x16x4_f32(false, a0, false, b0,
                                                    (short)0, c00, false, false);
        c01 = __builtin_amdgcn_wmma_f32_16x16x4_f32(false, a0, false, b1,
                                                    (short)0, c01, false, false);
        c10 = __builtin_amdgcn_wmma_f32_16x16x4_f32(false, a1, false, b0,
                                                    (short)0, c10, false, false);
        c11 = __builtin_amdgcn_wmma_f32_16x16x4_f32(false, a1, false, b1,
                                                    (short)0, c11, false, false);
    }

#pragma unroll
    for (int r = 0; r < 8; ++r) {
        const int row0 = rowBase + 8 * h + r;
        const int row1 = row0 + 16;
        const int col0 = colBase + l;
        const int col1 = col0 + 16;
        const float vals[4] = {c00[r], c01[r], c10[r], c11[r]};
        const int rows[4] = {row0, row0, row1, row1};
        const int cols[4] = {col0, col1, col0, col1};
#pragma unroll
        for (int q = 0; q < 4; ++q) {
            const size_t o = (size_t)rows[q] * ldc + cols[q];
            const float acc = vals[q];
            float e;
            if (mode == 0)      e = acc;
            else if (mode == 1) e = acc - aux1[cols[q]];
            else if (mode == 2) e = aux1[o] - acc;
            else                e = aux1[o] - (s1 * acc + s2 * aux2[o]);
            Cout[o] = e;
        }
    }
}

// --------------------------- small kernels ---------------------------------

// Augmented [L | I] for Gauss-Jordan (1024 x 2048)
__global__ void k_init_aug(float* __restrict__ aug, const float* __restrict__ L)
{
    int idx = blockIdx.x * blockDim.x + threadIdx.x;
    if (idx >= D_DIM * 2 * D_DIM) return;
    int i = idx / (2 * D_DIM), j = idx % (2 * D_DIM);
    aug[idx] = (j < D_DIM) ? L[(size_t)i * D_DIM + j]
                           : ((j - D_DIM == i) ? 1.0f : 0.0f);
}

// One GJ elimination step for pivot k; updates only columns j > k
// (columns < k are stale-but-dead), so no write/read hazards exist.
__global__ void k_gj_step(float* __restrict__ aug, int k)
{
    int j = k + 1 + blockIdx.x * blockDim.x + threadIdx.x;
    int i = blockIdx.y;
    if (j >= 2 * D_DIM || i == k) return;
    float piv = aug[(size_t)k * 2 * D_DIM + k];
    float f = aug[(size_t)i * 2 * D_DIM + k] / piv;
    aug[(size_t)i * 2 * D_DIM + j] -= f * aug[(size_t)k * 2 * D_DIM + j];
}

// LinvT[j*D + k] = Linv[k][j] = aug[k][D+j] / aug[k][k]
__global__ void k_extract_invT(float* __restrict__ linvT,
                               const float* __restrict__ aug)
{
    int idx = blockIdx.x * blockDim.x + threadIdx.x;
    if (idx >= D_DIM * D_DIM) return;
    int k = idx % D_DIM, j = idx / D_DIM;
    linvT[idx] = aug[(size_t)k * 2 * D_DIM + D_DIM + j] /
                 aug[(size_t)k * 2 * D_DIM + k];
}

// Lscaled[i,k] = L[i,k] * Lam[k]
__global__ void k_scaleL(float* __restrict__ lsc, const float* __restrict__ L,
                         const float* __restrict__ lam)
{
    int idx = blockIdx.x * blockDim.x + threadIdx.x;
    if (idx >= D_DIM * D_DIM) return;
    lsc[idx] = L[idx] * lam[idx % D_DIM];
}

// round 0 shortcut: x = -rho0 * c
__global__ void k_negscale(float* __restrict__ x, const float* __restrict__ c)
{
    int idx = blockIdx.x * blockDim.x + threadIdx.x;
    if (idx >= NELEM) return;
    x[idx] = -3.0f * c[idx];
}

// Dykstra init: p=q=0, yout=x0, done=0
__global__ void k_dyk_init(float* __restrict__ p, float* __restrict__ q,
                           float* __restrict__ yout, const float* __restrict__ x0,
                           unsigned* __restrict__ flags)
{
    int idx = blockIdx.x * blockDim.x + threadIdx.x;
    if (idx >= NELEM) return;
    p[idx] = 0.0f; q[idx] = 0.0f; yout[idx] = x0[idx];
    if (idx == 0) flags[1] = 0u;
}

// t = x + p ; also reset err accumulator
__global__ void k_t(float* __restrict__ t, const float* __restrict__ x,
                    const float* __restrict__ p, unsigned* __restrict__ flags)
{
    int idx = blockIdx.x * blockDim.x + threadIdx.x;
    if (idx >= NELEM) return;
    t[idx] = x[idx] + p[idx];
    if (idx == 0) flags[0] = 0u;   // err bits (non-negative float)
}

// block-reduce max(relu(-y)) -> atomicMax on float bits (values >= 0)
__global__ void k_err(const float* __restrict__ y, unsigned* __restrict__ flags)
{
    __shared__ float sm[256];
    int idx = blockIdx.x * blockDim.x + threadIdx.x;
    float v = (idx < NELEM) ? fmaxf(0.0f, -y[idx]) : 0.0f;
    sm[threadIdx.x] = v;
    __syncthreads();
    for (int s = 128; s > 0; s >>= 1) {
        if ((int)threadIdx.x < s)
            sm[threadIdx.x] = fmaxf(sm[threadIdx.x], sm[threadIdx.x + s]);
        __syncthreads();
    }
    if (threadIdx.x == 0) atomicMax(&flags[0], __float_as_uint(sm[0]));
}

// if !done(prev): yout=y; x=relu(y+q); p=t-y; q=y+q-x
__global__ void k_finalize(const unsigned* __restrict__ flags,
                           const float* __restrict__ y, float* __restrict__ yout,
                           const float* __restrict__ t, float* __restrict__ x,
                           float* __restrict__ p, float* __restrict__ q)
{
    if (flags[1]) return;
    int idx = blockIdx.x * blockDim.x + threadIdx.x;
    if (idx >= NELEM) return;
    float yv = y[idx], qv = q[idx];
    yout[idx] = yv;
    float xn = fmaxf(yv + qv, 0.0f);
    p[idx] = t[idx] - yv;
    q[idx] = yv + qv - xn;
    x[idx] = xn;
}

// done |= (err < TOL)   (runs after k_finalize consumed done_prev)
__global__ void k_done(unsigned* __restrict__ flags)
{
    if (blockIdx.x == 0 && threadIdx.x == 0) {
        if (!flags[1] && __uint_as_float(flags[0]) < TOLV) flags[1] = 1u;
    }
}

// --------------------------- host orchestration ----------------------------

static inline void gemm_nt(hipStream_t s, const float* A, int lda,
                           const float* B, int ldb, float* C, int ldc,
                           int M, int N, int K, int mode,
                           const float* aux1, const float* aux2,
                           float s1, float s2)
{
    dim3 g(N / 64, M / 64), b(128);
    k_gemm_nt<<<g, b, 0, s>>>(A, B, C, K, lda, ldb, ldc, mode, aux1, aux2, s1, s2);
}

extern "C" void kernel_launch(void* const* d_in, const int* in_sizes, int n_in,
                              void* d_out, int out_size, void* d_ws, size_t ws_size,
                              hipStream_t stream)
{
    const float* c   = (const float*)d_in[0];  // [512,1024]
    const float* A   = (const float*)d_in[1];  // [256,1024]
    const float* b   = (const float*)d_in[2];  // [256]
    const float* AA  = (const float*)d_in[3];  // [1024,256]
    const float* L   = (const float*)d_in[4];  // [1024,1024]
    const float* Lam = (const float*)d_in[5];  // [1024]

    char* ws = (char*)d_ws;
    const size_t MB = 1024 * 1024;
    float* aug    = (float*)(ws + 0);        // 8 MB  [1024 x 2048]
    float* lsc    = (float*)(ws + 8  * MB);  // 4 MB
    float* linvT  = (float*)(ws + 12 * MB);  // 4 MB
    float* Mmat   = (float*)(ws + 16 * MB);  // 4 MB
    float* xA     = (float*)(ws + 20 * MB);  // 2 MB
    float* xB     = (float*)(ws + 22 * MB);  // 2 MB
    float* pbuf   = (float*)(ws + 24 * MB);  // 2 MB
    float* qbuf   = (float*)(ws + 26 * MB);  // 2 MB
    float* tbuf   = (float*)(ws + 28 * MB);  // 2 MB
    float* ybuf   = (float*)(ws + 30 * MB);  // 2 MB
    float* youtW  = (float*)(ws + 32 * MB);  // 2 MB
    float* Sbuf   = (float*)(ws + 34 * MB);  // 0.5 MB [512 x 256]
    unsigned* flg = (unsigned*)(ws + 35 * MB); // [0]=err bits, [1]=done

    const int EBLK = 256;
    const int EW_N  = (NELEM + EBLK - 1) / EBLK;       // 2048 blocks
    const int DD_N  = (D_DIM * D_DIM + EBLK - 1) / EBLK;
    const int AUG_N = (D_DIM * 2 * D_DIM + EBLK - 1) / EBLK;

    // ---- M = (L * Lam[None,:]) @ inv(L) ----
    k_init_aug<<<AUG_N, EBLK, 0, stream>>>(aug, L);
    k_scaleL<<<DD_N, EBLK, 0, stream>>>(lsc, L, Lam);
    for (int k = 0; k < D_DIM; ++k) {
        int width = 2 * D_DIM - (k + 1);
        dim3 g((width + EBLK - 1) / EBLK, D_DIM);
        k_gj_step<<<g, EBLK, 0, stream>>>(aug, k);
    }
    k_extract_invT<<<DD_N, EBLK, 0, stream>>>(linvT, aug);
    // Mmat[i,j] = sum_k lsc[i,k] * linvT[j,k]   (NT)
    gemm_nt(stream, lsc, D_DIM, linvT, D_DIM, Mmat, D_DIM,
            D_DIM, D_DIM, D_DIM, 0, nullptr, nullptr, 0.f, 0.f);

    // ---- rounds ----
    const float rho = 3.0f, xrho = 0.5f;   // FACTOR == 1.0
    float* xbuf[2] = {xA, xB};
    int nextw = 1;
    const float* xcur;

    // round 0: x = 0 - (xrho*0 + rho*c) = -rho*c
    k_negscale<<<EW_N, EBLK, 0, stream>>>(xA, c);
    xcur = xA;

    for (int round = 0; round < 3; ++round) {
        if (round > 0) {
            float* xn = xbuf[nextw];
            // xn = xcur - (xrho * (xcur @ M^T) + rho * c)
            gemm_nt(stream, xcur, D_DIM, Mmat, D_DIM, xn, D_DIM,
                    B_DIM, D_DIM, D_DIM, 3, xcur, c, xrho, rho);
            xcur = xn;
            nextw ^= 1;
        }
        float* x = (float*)xcur;                   // Dykstra-internal state
        float* yout = (round == 2) ? (float*)d_out : youtW;

        k_dyk_init<<<EW_N, EBLK, 0, stream>>>(pbuf, qbuf, yout, x, flg);
        for (int it = 0; it < 30; ++it) {
            k_t<<<EW_N, EBLK, 0, stream>>>(tbuf, x, pbuf, flg);
            // S = t @ A^T - b
            gemm_nt(stream, tbuf, D_DIM, A, D_DIM, Sbuf, MC_DIM,
                    B_DIM, MC_DIM, D_DIM, 1, b, nullptr, 0.f, 0.f);
            // y = t - S @ AA^T
            gemm_nt(stream, Sbuf, MC_DIM, AA, MC_DIM, ybuf, D_DIM,
                    B_DIM, D_DIM, MC_DIM, 2, tbuf, nullptr, 0.f, 0.f);
            k_err<<<EW_N, EBLK, 0, stream>>>(ybuf, flg);
            k_finalize<<<EW_N, EBLK, 0, stream>>>(flg, ybuf, yout, tbuf,
                                                  x, pbuf, qbuf);
            k_done<<<1, 32, 0, stream>>>(flg);
        }
        xcur = yout;   // x = project_pol(x)
    }
}